// MultiHeadAttention_89472758710958
// MI455X (gfx1250) — compile-verified
//
#include <hip/hip_runtime.h>
#include <hip/hip_bf16.h>
#include <stdint.h>

typedef __attribute__((ext_vector_type(2))) float v2f;
typedef __attribute__((ext_vector_type(8))) float v8f;
typedef __attribute__((ext_vector_type(4))) uint32_t u32x4;
typedef __attribute__((ext_vector_type(8))) uint32_t u32x8;

// ---- CDNA5 async memory->LDS (ASYNCcnt path, CDNA5 ISA) ----
__device__ __forceinline__ void async_ld_b128(void* lds, const void* gptr) {
  uint32_t loff = (uint32_t)(uintptr_t)lds;   // low 32 bits = LDS offset
  asm volatile("global_load_async_to_lds_b128 %0, %1, off"
               :: "v"(loff), "v"(gptr)
               : "memory");
}
__device__ __forceinline__ void wait_async0() {
  asm volatile("s_wait_asynccnt 0x0" ::: "memory");
}
__device__ __forceinline__ void wait_ds0() {
  asm volatile("s_wait_dscnt 0x0" ::: "memory");
}

// ---- CDNA5 Tensor Data Mover: 2D tile, global -> LDS (TENSORcnt path) ----
// D# group0: [1:0]=count=1, [63:32]=lds_addr, [120:64]=global_addr, [127:126]=type(2)
// D# group1: data_size=4B, LDS row padding via pad_interval/pad_amount,
//            tensor dims set "large" (tiles never clip), dim0 stride in elems.
__device__ __forceinline__ void tdm_load_2d(uint32_t lds_addr, const void* gaddr,
                                            uint32_t tile_cols, uint32_t tile_rows,
                                            uint32_t stride_elems,
                                            uint32_t pad_interval, uint32_t pad_amount) {
  uint64_t ga = (uint64_t)(uintptr_t)gaddr;
  u32x4 g0;
  g0[0] = 1u;                                   // count=1 (valid), user mode
  g0[1] = lds_addr;                             // LDS byte address
  g0[2] = (uint32_t)ga;                         // global_addr[31:0]
  g0[3] = ((uint32_t)(ga >> 32) & 0x01FFFFFFu)  // global_addr[56:32]
          | 0x80000000u;                        // type = 2 ("image")
  u32x8 g1;
  g1[0] = (2u << 16)                            // data_size = 4 bytes
        | (1u << 20)                            // pad_enable
        | (pad_interval << 22)                  // pad every 2^(i+1)*... dwords
        | (pad_amount << 25);                   // pad amount (code N -> N+1 dwords)
  g1[1] = 0u;                                   // atomic_barrier=0, tensor_dim0[15:0]=0
  g1[2] = 0x10u;                                // tensor_dim0 = 1<<20, tensor_dim1[15:0]=0
  g1[3] = 0x10u | (tile_cols << 16);            // tensor_dim1 = 1<<20, tile_dim0
  g1[4] = tile_rows;                            // tile_dim1 (tile_dim2 = 0)
  g1[5] = stride_elems;                         // tensor_dim0_stride[31:0]
  g1[6] = 0u;                                   // stride hi, tensor_dim1_stride lo
  g1[7] = 0u;
  asm volatile("tensor_load_to_lds %0, %1" :: "s"(g0), "s"(g1) : "memory");
}

// -------------------------------------------------------------------------
// Kernel 1: kqv[M=B*T=32768, 1536] = x[M, 512] @ W[512, 1536]   (all fp32)
// Block: 256 threads = 8 waves (2 M-waves x 4 N-waves), tile 64(M) x 128(N)
// Double-buffered LDS tiles staged by the Tensor Data Mover (wave 0 issues
// one tensor_load_to_lds per tile); WMMA f32 16x16x4 compute.
// -------------------------------------------------------------------------
__global__ __launch_bounds__(256) void qkv_gemm_f32wmma(
    const float* __restrict__ x, const float* __restrict__ w,
    float* __restrict__ kqv) {
  __shared__ float As[2][64 * 20];    // 64x16 tile, TDM-padded rows of 20
  __shared__ float Bs[2][16 * 132];   // 16x128 tile, TDM-padded rows of 132

  const int tid  = threadIdx.x;
  const int lane = tid & 31;
  const int wave = tid >> 5;
  const int wm   = wave >> 2;      // 0..1  (32 rows each)
  const int wn   = wave & 3;       // 0..3  (32 cols each)
  const int lh   = lane >> 4;
  const int ll   = lane & 15;
  const int m0   = blockIdx.x * 64;
  const int n0   = blockIdx.y * 128;

  v8f acc[2][2] = {};

  // A rows: 16 floats (64B) -> pad_interval=3 (64B), pad 4 dwords -> pitch 20
  // B rows: 128 floats (512B) -> pad_interval=6 (512B), pad 4 dwords -> pitch 132
  auto prefetch = [&](int buf, int k0) {
    tdm_load_2d((uint32_t)(uintptr_t)&As[buf][0],
                x + (size_t)m0 * 512 + k0, /*cols*/16, /*rows*/64,
                /*stride*/512, /*pad_i*/3, /*pad_a*/3);
    tdm_load_2d((uint32_t)(uintptr_t)&Bs[buf][0],
                w + (size_t)k0 * 1536 + n0, /*cols*/128, /*rows*/16,
                /*stride*/1536, /*pad_i*/6, /*pad_a*/3);
  };

  if (wave == 0) {
    prefetch(0, 0);
    __builtin_amdgcn_s_wait_tensorcnt(0);
  }
  __syncthreads();

  for (int k0 = 0; k0 < 512; k0 += 16) {
    const int cur = (k0 >> 4) & 1;
    if (wave == 0 && k0 + 16 < 512)
      prefetch(cur ^ 1, k0 + 16);        // TDM DMA overlaps WMMAs below

#pragma unroll
    for (int ks = 0; ks < 4; ++ks) {
      const int kk = ks * 4 + lh * 2;
      v2f a[2], b[2];
#pragma unroll
      for (int ms = 0; ms < 2; ++ms)
        a[ms] = *(const v2f*)(&As[cur][(wm * 32 + ms * 16 + ll) * 20 + kk]);
#pragma unroll
      for (int ns = 0; ns < 2; ++ns) {
        const int col = wn * 32 + ns * 16 + ll;
        v2f t;
        t.x = Bs[cur][kk * 132 + col];
        t.y = Bs[cur][(kk + 1) * 132 + col];
        b[ns] = t;
      }
#pragma unroll
      for (int ms = 0; ms < 2; ++ms)
#pragma unroll
        for (int ns = 0; ns < 2; ++ns)
          acc[ms][ns] = __builtin_amdgcn_wmma_f32_16x16x4_f32(
              false, a[ms], false, b[ns], (short)0, acc[ms][ns], false, false);
    }
    if (wave == 0) __builtin_amdgcn_s_wait_tensorcnt(0);  // next tile landed
    __syncthreads();   // readers done with cur; TDM data visible to all
  }

#pragma unroll
  for (int ms = 0; ms < 2; ++ms)
#pragma unroll
    for (int ns = 0; ns < 2; ++ns)
#pragma unroll
      for (int r = 0; r < 8; ++r) {
        const int row = m0 + wm * 32 + ms * 16 + r + lh * 8;
        const int col = n0 + wn * 32 + ns * 16 + ll;
        kqv[(size_t)row * 1536 + col] = acc[ms][ns][r];
      }
}

// -------------------------------------------------------------------------
// Kernel 2: flash attention, fp32 WMMA.
// Block = 128 threads (4 waves) per (batch b, 16-query tile qt).
// kqv row layout (ld=1536): [0,512)=k  [512,1024)=q  [1024,1536)=v
// Wave w: partial S over C-slice [w*128,(w+1)*128) (LDS-reduced), owns
// output cols [w*128,(w+1)*128). V slice async-DMA'd to LDS (ASYNCcnt),
// overlapped with the S WMMA block + softmax.
// -------------------------------------------------------------------------
__global__ __launch_bounds__(128) void flash_attn_f32wmma(
    const float* __restrict__ kqv, float* __restrict__ out) {
  __shared__ float Ssh[4][256];        // per-wave partial S (D-layout order)
  __shared__ float Psh[4][256];        // per-wave P, row-major 16x16
  __shared__ float Vsh[4][16 * 132];   // per-wave V slice 16x128 (padded)

  const int tid  = threadIdx.x;
  const int lane = tid & 31;
  const int w    = tid >> 5;
  const int lh   = lane >> 4, ll = lane & 15;
  const int qt   = blockIdx.x;          // 0..63 (query tile)
  const int b    = blockIdx.y;          // 0..31 (batch)
  const size_t rowbase = (size_t)b * 1024;
  const float  scale   = 0.044194173824159216f;  // 512^-0.5
  const int    cw      = w * 128;

  v8f   U[8] = {};
  float m_s[8], l_s[8];
#pragma unroll
  for (int r = 0; r < 8; ++r) { m_s[r] = -1e30f; l_s[r] = 0.0f; }

  const float* qrow = kqv + (rowbase + (size_t)qt * 16 + ll) * 1536 + 512;

  for (int s = 0; s <= qt; ++s) {
    // WAR guard: previous step's ds reads of this wave's Vsh must retire
    // before the async engine overwrites it.
    wait_ds0();
    // ---- issue V-slice DMA now; it hides under S WMMAs + softmax ----
    const float* vbase = kqv + (rowbase + (size_t)s * 16) * 1536 + 1024 + cw;
#pragma unroll
    for (int i = 0; i < 16; ++i)
      async_ld_b128(&Vsh[w][i * 132 + lane * 4],
                    vbase + (size_t)i * 1536 + lane * 4);

    // ---- partial S = Q Kt over this wave's C-slice ----
    v8f Sacc = {};
    const float* krow = kqv + (rowbase + (size_t)s * 16 + ll) * 1536;
#pragma unroll 8
    for (int ks = 0; ks < 32; ++ks) {
      const int c = cw + ks * 4 + lh * 2;
      v2f a  = *(const v2f*)(qrow + c);
      v2f bb = *(const v2f*)(krow + c);
      Sacc = __builtin_amdgcn_wmma_f32_16x16x4_f32(
          false, a, false, bb, (short)0, Sacc, false, false);
    }

    // ---- reduce partial S across the 4 waves via LDS ----
#pragma unroll
    for (int r = 0; r < 8; ++r) Ssh[w][r * 32 + lane] = Sacc[r];
    __syncthreads();
    float Sv[8];
#pragma unroll
    for (int r = 0; r < 8; ++r)
      Sv[r] = (Ssh[0][r * 32 + lane] + Ssh[1][r * 32 + lane]) +
              (Ssh[2][r * 32 + lane] + Ssh[3][r * 32 + lane]);
    __syncthreads();

    // ---- scale, causal mask, online softmax update ----
#pragma unroll
    for (int r = 0; r < 8; ++r) {
      float sv = Sv[r] * scale;
      const int qi = qt * 16 + r + lh * 8;  // query row of this element
      const int ki = s * 16 + ll;           // key col of this element
      Sv[r] = (ki <= qi) ? sv : -1e30f;
    }
#pragma unroll
    for (int r = 0; r < 8; ++r) {
      float mx = Sv[r];
      mx = fmaxf(mx, __shfl_xor(mx, 1));
      mx = fmaxf(mx, __shfl_xor(mx, 2));
      mx = fmaxf(mx, __shfl_xor(mx, 4));
      mx = fmaxf(mx, __shfl_xor(mx, 8));
      const float mnew  = fmaxf(m_s[r], mx);
      const float alpha = __expf(m_s[r] - mnew);
      const float pv    = __expf(Sv[r] - mnew);
      float sum = pv;
      sum += __shfl_xor(sum, 1);
      sum += __shfl_xor(sum, 2);
      sum += __shfl_xor(sum, 4);
      sum += __shfl_xor(sum, 8);
      l_s[r] = l_s[r] * alpha + sum;
      m_s[r] = mnew;
#pragma unroll
      for (int nt = 0; nt < 8; ++nt) U[nt][r] *= alpha;
      // D-layout -> row-major P staging (wave-private buffer)
      Psh[w][(r + lh * 8) * 16 + ll] = pv;
    }

    // ---- V DMA must be complete before B-frag reads ----
    wait_async0();

    // ---- U += P(16x16) x V(16 x 128-slice) ----
#pragma unroll
    for (int ks = 0; ks < 4; ++ks) {
      const int kd = ks * 4 + lh * 2;
      v2f a = *(const v2f*)(&Psh[w][ll * 16 + kd]);
#pragma unroll
      for (int nt = 0; nt < 8; ++nt) {
        v2f bb;
        bb.x = Vsh[w][kd * 132 + nt * 16 + ll];
        bb.y = Vsh[w][(kd + 1) * 132 + nt * 16 + ll];
        U[nt] = __builtin_amdgcn_wmma_f32_16x16x4_f32(
            false, a, false, bb, (short)0, U[nt], false, false);
      }
    }
  }

  // ---- finalize: out = U / l ----
#pragma unroll
  for (int r = 0; r < 8; ++r) {
    const float inv = 1.0f / l_s[r];
    const int   row = qt * 16 + r + lh * 8;
    float* orow = out + (rowbase + row) * 512 + cw;
#pragma unroll
    for (int nt = 0; nt < 8; ++nt) orow[nt * 16 + ll] = U[nt][r] * inv;
  }
}

// -------------------------------------------------------------------------
extern "C" void kernel_launch(void* const* d_in, const int* in_sizes, int n_in,
                              void* d_out, int out_size, void* d_ws, size_t ws_size,
                              hipStream_t stream) {
  const float* x = (const float*)d_in[0];   // [32,1024,512]
  const float* w = (const float*)d_in[1];   // [512,1536]
  float* outp = (float*)d_out;              // [32,1024,512]
  float* kqv  = (float*)d_ws;               // 32768*1536 fp32 = 192 MB scratch

  dim3 g1(512, 12), b1(256);                // (M/64, 1536/128)
  qkv_gemm_f32wmma<<<g1, b1, 0, stream>>>(x, w, kqv);

  dim3 g2(64, 32), b2(128);                 // (T/16, B)
  flash_attn_f32wmma<<<g2, b2, 0, stream>>>(kqv, outp);
}